// WaveletParsingLayer_78769700208779
// MI455X (gfx1250) — compile-verified
//
#include <hip/hip_runtime.h>
#include <stdint.h>

// ---------------------------------------------------------------------------
// Problem constants (static from the reference: B=64, R=8, L=65536).
// ---------------------------------------------------------------------------
namespace {
constexpr int kB      = 64;
constexpr int kR      = 8;
constexpr int kL      = 65536;
constexpr int kKRow   = kL / 2;          // designed valid per (b,r) segment
constexpr int kOutLen = kR * kKRow;      // 262144 per batch row
constexpr unsigned kFillerBits = 0x3F8CCCCDu;  // bit pattern of 1.1f
constexpr float    kFiller     = 1.1f;

constexpr int TPB   = 256;               // 8 wave32 waves per block
constexpr int WAVES = TPB / 32;
constexpr int VEC   = 4;                 // float4 per thread per tile
constexpr int TILE  = TPB * VEC;         // 1024 elements / tile
constexpr int ITERS = kL / TILE;         // 64 tiles per segment

constexpr int ABUF  = 4;                 // async LDS ring depth (16 KB)
} // namespace

// gfx1250 async global->LDS copy (GV mode), tracked with ASYNCcnt.
// dst = wave-relative LDS byte address, g = 64-bit global address.
__device__ __forceinline__ void async_copy_b128(unsigned dst, const float4* g)
{
    asm volatile("global_load_async_to_lds_b128 %0, %1, off"
                 :: "v"(dst), "v"(g)
                 : "memory");
}

// ---------------------------------------------------------------------------
// Phase 1: per-segment non-filler counts. The async engine streams 128-bit
// tiles into a 4-deep LDS ring (ASYNCcnt) while the wave counts the previous
// tile — deep HBM prefetch at zero VGPR cost. Also warms the 192MB L2 so the
// phase-3 re-read is effectively free of HBM traffic.
// ---------------------------------------------------------------------------
__global__ __launch_bounds__(TPB)
void seg_count_kernel(const float4* __restrict__ x3, unsigned* __restrict__ counts)
{
    const int tid = threadIdx.x;
    const int seg = blockIdx.x;
    const float4* __restrict__ gp = x3 + (size_t)seg * (kL / 4) + tid;

    __shared__ float4 s_tile[ABUF][TPB];             // 16 KB staging ring

    // Prime the pipeline: ABUF tiles in flight per wave.
#pragma unroll
    for (int k = 0; k < ABUF; ++k) {
        const unsigned dst = (unsigned)(uintptr_t)&s_tile[k][tid];
        async_copy_b128(dst, gp + k * TPB);
    }

    unsigned c = 0;
    for (int it = 0; it < ITERS - ABUF; ++it) {
        // Oldest outstanding tile (ours: each lane reads only its own slot,
        // written by its own wave's async op; loads complete in order).
        asm volatile("s_wait_asynccnt %0" :: "i"(ABUF - 1) : "memory");
        const int buf = it & (ABUF - 1);
        const float4 v = s_tile[buf][tid];
        c += (__float_as_uint(v.x) != kFillerBits);
        c += (__float_as_uint(v.y) != kFillerBits);
        c += (__float_as_uint(v.z) != kFillerBits);
        c += (__float_as_uint(v.w) != kFillerBits);
        // Close the ds-read -> async-overwrite hazard, then refill the slot.
        asm volatile("s_wait_dscnt 0x0" ::: "memory");
        const unsigned dst = (unsigned)(uintptr_t)&s_tile[buf][tid];
        async_copy_b128(dst, gp + (it + ABUF) * TPB);
    }

    // Drain the tail of the ring.
    asm volatile("s_wait_asynccnt 0x0" ::: "memory");
#pragma unroll
    for (int it = ITERS - ABUF; it < ITERS; ++it) {
        const float4 v = s_tile[it & (ABUF - 1)][tid];
        c += (__float_as_uint(v.x) != kFillerBits);
        c += (__float_as_uint(v.y) != kFillerBits);
        c += (__float_as_uint(v.z) != kFillerBits);
        c += (__float_as_uint(v.w) != kFillerBits);
    }

    __shared__ unsigned s_sum;
    if (tid == 0) s_sum = 0u;
    __syncthreads();
    atomicAdd(&s_sum, c);
    __syncthreads();
    if (tid == 0) counts[seg] = s_sum;
}

// ---------------------------------------------------------------------------
// Phase 2: exclusive prefix of the 8 segment counts inside each batch row.
// ---------------------------------------------------------------------------
__global__ __launch_bounds__(512)
void seg_scan_kernel(const unsigned* __restrict__ counts,
                     unsigned* __restrict__ bases,
                     unsigned* __restrict__ totals)
{
    const int t = threadIdx.x;
    if (t >= kB * kR) return;
    const int b = t >> 3;
    const int r = t & 7;
    unsigned base = 0;
    for (int k = 0; k < r; ++k) base += counts[(b << 3) + k];
    bases[t] = base;
    if (r == kR - 1) totals[b] = base + counts[t];
}

// ---------------------------------------------------------------------------
// Phase 3: ordered compaction. One block per segment; per 1024-element tile:
// wave32 ballot scan (4 slot ballots + popc), cross-wave scan via parity-
// double-buffered LDS (single barrier per tile), dense non-temporal stores.
// Loads mostly hit L2 (warmed by phase 1).
// ---------------------------------------------------------------------------
__global__ __launch_bounds__(TPB)
void seg_compact_kernel(const float4* __restrict__ x3,
                        const unsigned* __restrict__ bases,
                        float* __restrict__ out)
{
    const int seg  = blockIdx.x;
    const int brow = seg >> 3;
    const float4* __restrict__ in4 = x3 + (size_t)seg * (kL / 4);

    const unsigned base  = bases[seg];
    float* __restrict__ ob = out + (size_t)brow * kOutLen + base;
    const unsigned limit = (unsigned)kOutLen - base;

    const int tid  = threadIdx.x;
    const int lane = tid & 31;
    const int wid  = tid >> 5;
    const unsigned lt = (1u << lane) - 1u;

    __shared__ unsigned s_wsum[2][WAVES];

    unsigned running = 0;
    float4 cur = in4[tid];                       // software-pipelined preload

    for (int it = 0; it < ITERS; ++it) {
        float4 nxt = cur;
        if (it + 1 < ITERS) nxt = in4[(it + 1) * TPB + tid];
        if (it + 8 < ITERS && (tid & 3) == 0)
            __builtin_prefetch(&in4[(it + 8) * TPB + tid], 0, 3);

        const bool f0 = __float_as_uint(cur.x) != kFillerBits;
        const bool f1 = __float_as_uint(cur.y) != kFillerBits;
        const bool f2 = __float_as_uint(cur.z) != kFillerBits;
        const bool f3 = __float_as_uint(cur.w) != kFillerBits;

        // Wave32 ballot scan: element order is (lane-major, slot-minor).
        const unsigned m0 = (unsigned)__ballot(f0);
        const unsigned m1 = (unsigned)__ballot(f1);
        const unsigned m2 = (unsigned)__ballot(f2);
        const unsigned m3 = (unsigned)__ballot(f3);

        const unsigned excl = __popc(m0 & lt) + __popc(m1 & lt) +
                              __popc(m2 & lt) + __popc(m3 & lt);
        const unsigned wtot = __popc(m0) + __popc(m1) + __popc(m2) + __popc(m3);

        const int par = it & 1;                  // parity double buffer -> 1 barrier
        if (lane == 0) s_wsum[par][wid] = wtot;
        __syncthreads();

        unsigned wbase = 0, btot = 0;
#pragma unroll
        for (int w = 0; w < WAVES; ++w) {
            const unsigned t = s_wsum[par][w];
            btot  += t;
            wbase += (w < wid) ? t : 0u;
        }

        unsigned p = running + wbase + excl;
        if (f0) { if (p < limit) __builtin_nontemporal_store(cur.x, ob + p); ++p; }
        if (f1) { if (p < limit) __builtin_nontemporal_store(cur.y, ob + p); ++p; }
        if (f2) { if (p < limit) __builtin_nontemporal_store(cur.z, ob + p); ++p; }
        if (f3) { if (p < limit) __builtin_nontemporal_store(cur.w, ob + p); ++p; }

        running += btot;
        cur = nxt;
    }
}

// ---------------------------------------------------------------------------
// Phase 4: tail fill with the filler value (reference pads with 1.1f when a
// valid sample collides with the filler bit pattern). Normally writes 0 elems.
// ---------------------------------------------------------------------------
__global__ __launch_bounds__(TPB)
void tail_fill_kernel(const unsigned* __restrict__ totals, float* __restrict__ out)
{
    const int b = blockIdx.x;
    const unsigned V = totals[b];
    float* __restrict__ row = out + (size_t)b * kOutLen;
    for (unsigned i = V + threadIdx.x; i < (unsigned)kOutLen; i += TPB)
        __builtin_nontemporal_store(kFiller, row + i);
}

// ---------------------------------------------------------------------------
// Host entry. Inputs: d_in[0]=x1 (unused), d_in[1]=x2 (unused), d_in[2]=x3,
// d_in[3]=out_len (static). Scratch: 512 counts + 512 bases + 64 totals.
// ---------------------------------------------------------------------------
extern "C" void kernel_launch(void* const* d_in, const int* in_sizes, int n_in,
                              void* d_out, int out_size, void* d_ws, size_t ws_size,
                              hipStream_t stream)
{
    (void)in_sizes; (void)n_in; (void)out_size; (void)ws_size;

    const float4* x3  = (const float4*)d_in[2];
    float*        out = (float*)d_out;
    unsigned*     ws  = (unsigned*)d_ws;

    unsigned* counts = ws;          // [512]
    unsigned* bases  = ws + 512;    // [512]
    unsigned* totals = ws + 1024;   // [64]

    seg_count_kernel  <<<kB * kR, TPB, 0, stream>>>(x3, counts);
    seg_scan_kernel   <<<1,       512, 0, stream>>>(counts, bases, totals);
    seg_compact_kernel<<<kB * kR, TPB, 0, stream>>>(x3, bases, out);
    tail_fill_kernel  <<<kB,      TPB, 0, stream>>>(totals, out);
}